// GCN_24962349924889
// MI455X (gfx1250) — compile-verified
//
#include <hip/hip_runtime.h>
#include <hip/hip_bf16.h>

typedef __attribute__((ext_vector_type(2))) float v2f;
typedef __attribute__((ext_vector_type(8))) float v8f;

#define D_FEAT 64

// ---------------------------------------------------------------------------
// Degree kernels: deg starts at 1.0 (self loop), count targets, then rsqrt.
// ---------------------------------------------------------------------------
__global__ void deg_init(float* __restrict__ deg, int n) {
    int i = blockIdx.x * blockDim.x + threadIdx.x;
    if (i < n) deg[i] = 1.0f;
}

__global__ void deg_count(const int* __restrict__ ei, float* __restrict__ deg, int E) {
    int e = blockIdx.x * blockDim.x + threadIdx.x;
    if (e < E) atomicAdd(&deg[ei[E + e]], 1.0f);   // col = targets
}

__global__ void deg_rsqrt(float* __restrict__ deg, int n) {
    int i = blockIdx.x * blockDim.x + threadIdx.x;
    if (i < n) deg[i] = rsqrtf(deg[i]);
}

// ---------------------------------------------------------------------------
// f32 GEMM via V_WMMA_F32_16X16X4_F32.
// C[N x 64] = A[N x 64] * W[64 x 64], both row-major.
// One block = 128 threads = 4 waves; block b computes rows [16b,16b+16);
// wave w computes output columns [16w, 16w+16). K looped 0..63 step 4.
// N must be a multiple of 16 (100000 = 6250*16). EXEC stays all-1s.
//
// ISA fragment layouts (wave32):
//   A 16x4 f32 : lane L (L<16) holds A[M=L,K=0..1] in v0,v1; lane L+16 holds K=2..3
//   B 4x16 f32 : lane L (L<16) holds B[K=0..1,N=L];          lane L+16 holds K=2..3
//   C/D 16x16  : vgpr v, lanes 0-15 -> D[M=v, N=lane]; lanes 16-31 -> D[M=8+v, N=lane-16]
// ---------------------------------------------------------------------------
__global__ void gemm64_wmma(const float* __restrict__ A,
                            const float* __restrict__ W,
                            float* __restrict__ out) {
    const int tileM = blockIdx.x;
    const int tileN = threadIdx.x >> 5;        // 0..3
    const int lane  = threadIdx.x & 31;
    const int m     = lane & 15;
    const int kb    = (lane >> 4) << 1;        // 0 or 2
    const int nn    = tileN * 16 + m;

    const float* arow = A + (size_t)(tileM * 16 + m) * D_FEAT;

    v8f c = {};
#pragma unroll
    for (int k0 = 0; k0 < D_FEAT; k0 += 4) {
        v2f a;
        a.x = arow[k0 + kb];
        a.y = arow[k0 + kb + 1];
        v2f b;
        b.x = W[(k0 + kb)     * D_FEAT + nn];
        b.y = W[(k0 + kb + 1) * D_FEAT + nn];
        c = __builtin_amdgcn_wmma_f32_16x16x4_f32(
                /*neg_a=*/false, a, /*neg_b=*/false, b,
                /*c_mod=*/(short)0, c, /*reuse_a=*/false, /*reuse_b=*/false);
    }

    const int mbase = tileM * 16 + ((lane >> 4) << 3);
#pragma unroll
    for (int v = 0; v < 8; ++v)
        out[(size_t)(mbase + v) * D_FEAT + nn] = c[v];
}

// ---------------------------------------------------------------------------
// out[i,f] = h[i,f] * dinv[i]^2 + bias[f]   (self-loop term + bias; INITIALIZES out)
// ---------------------------------------------------------------------------
__global__ void selfloop_bias_init(const float* __restrict__ h,
                                   const float* __restrict__ dinv,
                                   const float* __restrict__ bias,
                                   float* __restrict__ out, int n) {
    int gid = blockIdx.x * blockDim.x + threadIdx.x;
    if (gid >= n * D_FEAT) return;
    int i = gid >> 6;
    int f = gid & 63;
    float d = dinv[i];
    out[gid] = h[gid] * d * d + bias[f];
}

// ---------------------------------------------------------------------------
// Edge scatter: one wave per edge, each lane handles 2 features (float2 gather,
// two f32 atomics). out[col] += h[row] * dinv[row]*dinv[col].
// ---------------------------------------------------------------------------
__global__ void edge_scatter(const float* __restrict__ h,
                             const int* __restrict__ ei,
                             const float* __restrict__ dinv,
                             float* __restrict__ out, int E) {
    int gid = blockIdx.x * blockDim.x + threadIdx.x;   // E*32 threads
    int e = gid >> 5;
    if (e >= E) return;
    int f2 = (gid & 31) * 2;
    int r = ei[e];
    int c = ei[E + e];
    float w = dinv[r] * dinv[c];
    const float2 hv = *reinterpret_cast<const float2*>(h + (size_t)r * D_FEAT + f2);
    float* op = out + (size_t)c * D_FEAT + f2;
    atomicAdd(op,     hv.x * w);
    atomicAdd(op + 1, hv.y * w);
}

// ---------------------------------------------------------------------------
// Pooling: sums[64*64] and cnt[64] live contiguously in ws.
// batch is sorted, so each 64-node chunk flushes very few atomics.
// ---------------------------------------------------------------------------
__global__ void pool_zero(float* __restrict__ sums) {
    int i = blockIdx.x * blockDim.x + threadIdx.x;
    if (i < 64 * 64 + 64) sums[i] = 0.0f;
}

__global__ void pool_accum(const float* __restrict__ h,
                           const int* __restrict__ batch,
                           float* __restrict__ sums,
                           float* __restrict__ cnt, int n) {
    int f  = threadIdx.x;            // 64 threads
    int i0 = blockIdx.x * 64;
    if (i0 >= n) return;
    int iend = i0 + 64; if (iend > n) iend = n;
    int g = batch[i0];
    float acc = 0.0f, c = 0.0f;
    for (int i = i0; i < iend; ++i) {
        int gi = batch[i];
        if (gi != g) {
            atomicAdd(&sums[g * 64 + f], acc);
            if (f == 0) atomicAdd(&cnt[g], c);
            g = gi; acc = 0.0f; c = 0.0f;
        }
        acc += h[(size_t)i * D_FEAT + f];
        c += 1.0f;
    }
    atomicAdd(&sums[g * 64 + f], acc);
    if (f == 0) atomicAdd(&cnt[g], c);
}

__global__ void pool_final(const float* __restrict__ sums,
                           const float* __restrict__ cnt,
                           float* __restrict__ out) {
    int gid = blockIdx.x * blockDim.x + threadIdx.x;
    if (gid < 64 * 64) out[gid] = sums[gid] / fmaxf(cnt[gid >> 6], 1.0f);
}

// ---------------------------------------------------------------------------
extern "C" void kernel_launch(void* const* d_in, const int* in_sizes, int n_in,
                              void* d_out, int out_size, void* d_ws, size_t ws_size,
                              hipStream_t stream) {
    const float* x     = (const float*)d_in[0];
    const int*   ei    = (const int*)  d_in[1];
    const int*   batch = (const int*)  d_in[2];
    const float* W1    = (const float*)d_in[3];
    const float* b1    = (const float*)d_in[4];
    const float* W2    = (const float*)d_in[5];
    const float* b2    = (const float*)d_in[6];

    const int N = in_sizes[0] / D_FEAT;     // 100000 (multiple of 16)
    const int E = in_sizes[1] / 2;          // 1000000

    // Workspace layout (floats): dinv[N] | hA[N*64] | hB[N*64] | sums[4096] | cnt[64]
    float* dinv = (float*)d_ws;
    float* hA   = dinv + (((size_t)N + 63) & ~(size_t)63);
    float* hB   = hA + (size_t)N * D_FEAT;
    float* sums = hB + (size_t)N * D_FEAT;
    float* cnt  = sums + 64 * 64;

    const int T = 256;

    // Degrees (with self loop) -> dinv
    deg_init <<<(N + T - 1) / T, T, 0, stream>>>(dinv, N);
    deg_count<<<(E + T - 1) / T, T, 0, stream>>>(ei, dinv, E);
    deg_rsqrt<<<(N + T - 1) / T, T, 0, stream>>>(dinv, N);

    // Layer 1: hA = x @ W1 ; hB = D^-1/2 (A+I) D^-1/2 hA + b1
    gemm64_wmma<<<N / 16, 128, 0, stream>>>(x, W1, hA);
    selfloop_bias_init<<<((size_t)N * D_FEAT + T - 1) / T, T, 0, stream>>>(hA, dinv, b1, hB, N);
    edge_scatter<<<((size_t)E * 32 + T - 1) / T, T, 0, stream>>>(hA, ei, dinv, hB, E);

    // Layer 2: hA = hB @ W2 ; hB = D^-1/2 (A+I) D^-1/2 hA + b2   (ping-pong)
    gemm64_wmma<<<N / 16, 128, 0, stream>>>(hB, W2, hA);
    selfloop_bias_init<<<((size_t)N * D_FEAT + T - 1) / T, T, 0, stream>>>(hA, dinv, b2, hB, N);
    edge_scatter<<<((size_t)E * 32 + T - 1) / T, T, 0, stream>>>(hA, ei, dinv, hB, E);

    // Global mean pool
    pool_zero <<<(64 * 64 + 64 + T - 1) / T, T, 0, stream>>>(sums);
    pool_accum<<<(N + 63) / 64, 64, 0, stream>>>(hB, batch, sums, cnt, N);
    pool_final<<<(64 * 64 + T - 1) / T, T, 0, stream>>>(sums, cnt, (float*)d_out);
}